// Delta_30958124270136
// MI455X (gfx1250) — compile-verified
//
#include <hip/hip_runtime.h>
#include <stdint.h>

// Delta differential encoding with residual carry + floor-quantization.
// x: [32, 2048, 512] f32, scan along T=512. 65536 independent rows.
// Memory-bound (268 MB total) -> async global->LDS staging (gfx1250 path),
// per-wave double buffering, coalesced loads AND stores via LDS transpose.

#define T_LEN   512
#define TC      32                 // chunk length along T (floats)
#define TCP     36                 // padded LDS row stride (floats, 144B: 16B aligned, conflict-light)
#define BLOCK   128                // threads per block == rows per block (4 waves)
#define NCHUNK  (T_LEN / TC)       // 16
#define INV_SCALE 0.015625f        // 1/64

#if __has_builtin(__builtin_amdgcn_global_load_async_to_lds_b128) && \
    __has_builtin(__builtin_amdgcn_s_wait_asynccnt)
#define USE_ASYNC 1
#else
#define USE_ASYNC 0
#pragma message("gfx1250 async-to-LDS builtins NOT available: falling back to sync copy")
#endif

typedef int v4i __attribute__((ext_vector_type(4)));
typedef __attribute__((address_space(1))) v4i gv4i;   // global (AS1) int4
typedef __attribute__((address_space(3))) v4i sv4i;   // LDS (AS3) int4

__device__ __forceinline__ void copy16_g2l(const float* g, float* l) {
#if USE_ASYNC
  // GLOBAL_LOAD_ASYNC_TO_LDS_B128: 16B per lane, memory -> LDS, ASYNCcnt-tracked
  __builtin_amdgcn_global_load_async_to_lds_b128(
      (gv4i*)(uintptr_t)g,
      (sv4i*)(uint32_t)(uintptr_t)l,
      /*imm offset*/0, /*cpol*/0);
#else
  *(float4*)l = *(const float4*)g;
#endif
}

__device__ __forceinline__ void wait_async_le8() {
#if USE_ASYNC
  __builtin_amdgcn_s_wait_asynccnt(8);   // previous chunk's 8 ops retired (in-order)
#endif
}
__device__ __forceinline__ void wait_async_le0() {
#if USE_ASYNC
  __builtin_amdgcn_s_wait_asynccnt(0);
#endif
}

__device__ __forceinline__ float dq_step(float xv, float thr, float& pre, float& res) {
  // exact reference arithmetic order: delta = (x - pre) + res
  float d = (xv - pre) + res;
  float y = (fabsf(d) >= thr) ? d : 0.0f;
  res = d - y;          // 0 when emitted, else full delta
  pre = xv;
  return floorf(y * 64.0f) * INV_SCALE;
}

__global__ __launch_bounds__(BLOCK)
void Delta_30958124270136_kernel(const float* __restrict__ x,
                                 const float* __restrict__ thr_p,
                                 float* __restrict__ out)
{
  // [2 buffers][128 rows][36 floats] input staging + [128][36] output staging
  __shared__ __align__(16) float lin[2][BLOCK * TCP];
  __shared__ __align__(16) float lout[BLOCK * TCP];

  const int tid  = threadIdx.x;
  const int wave = tid >> 5;
  const int lane = tid & 31;
  const long rowBase = (long)blockIdx.x * BLOCK;

  const float thr = fmaxf(thr_p[0], INV_SCALE);

  // Copy assignment (per wave, fully self-contained -> no block barriers):
  // 8 lanes per row-chunk (32 floats = 8 x 16B), 4 rows per pass, 8 passes.
  const int rsub = lane >> 3;        // row within pass: 0..3
  const int csub = (lane & 7) * 4;   // float column within chunk: 0,4,...,28

  auto issue_chunk = [&](int k, int buf) {
    const int t0 = k * TC;
#pragma unroll
    for (int p = 0; p < 8; ++p) {
      const int r = (wave << 5) + 4 * p + rsub;          // row within block
      const float* g = x + (rowBase + r) * T_LEN + t0 + csub;
      copy16_g2l(g, &lin[buf][r * TCP + csub]);
    }
  };

  issue_chunk(0, 0);

  float pre = 0.0f, res = 0.0f;

  for (int k = 0; k < NCHUNK; ++k) {
    const int buf = k & 1;
    if (k + 1 < NCHUNK) {
      issue_chunk(k + 1, buf ^ 1);   // prefetch next chunk into other buffer
      wait_async_le8();              // chunk k (first 8 in-flight ops) has landed
    } else {
      wait_async_le0();
    }

    // Serial scan of this thread's row chunk (ds_load_b128 every 4 steps).
#pragma unroll
    for (int j = 0; j < TC / 4; ++j) {
      const float4 v = *(const float4*)&lin[buf][tid * TCP + 4 * j];
      float4 o;
      o.x = dq_step(v.x, thr, pre, res);
      o.y = dq_step(v.y, thr, pre, res);
      o.z = dq_step(v.z, thr, pre, res);
      o.w = dq_step(v.w, thr, pre, res);
      *(float4*)&lout[tid * TCP + 4 * j] = o;
    }

    // Coalesced write-out: same wave-local transpose pattern as the loads.
    const int t0 = k * TC;
#pragma unroll
    for (int p = 0; p < 8; ++p) {
      const int r = (wave << 5) + 4 * p + rsub;
      const float4 v = *(const float4*)&lout[r * TCP + csub];
      *(float4*)&out[(rowBase + r) * T_LEN + t0 + csub] = v;
    }
  }
}

extern "C" void kernel_launch(void* const* d_in, const int* in_sizes, int n_in,
                              void* d_out, int out_size, void* d_ws, size_t ws_size,
                              hipStream_t stream) {
  const float* x   = (const float*)d_in[0];   // [32, 2048, 512] f32
  const float* thr = (const float*)d_in[1];   // [1] f32
  float* out = (float*)d_out;                 // [32, 2048, 512] f32

  const int rows = in_sizes[0] / T_LEN;       // 65536
  const int grid = rows / BLOCK;              // 512 blocks x 128 threads (4 waves)

  Delta_30958124270136_kernel<<<dim3(grid), dim3(BLOCK), 0, stream>>>(x, thr, out);
}